// ComplexRenderer_7962869366961
// MI455X (gfx1250) — compile-verified
//
#include <hip/hip_runtime.h>
#include <math.h>

typedef float v2f __attribute__((ext_vector_type(2)));
typedef float v8f __attribute__((ext_vector_type(8)));

#define N_POINTS 32768
#define N_PRIMS  2048
#define NEG_HALF_LOG2E (-0.7213475204444817f)   // -0.5 * log2(e)

// ---------------------------------------------------------------------------
// Prep: per-primitive K=8 feature rows for the two fused GEMMs.
// featM is pre-scaled by -0.5*log2(e) so the maha-GEMM directly produces the
// exp2() argument (amplitude folded in log domain as well):
//   featM[p] = lam * [iv0, iv1, iv2, -2mu0*iv0, -2mu1*iv1, -2mu2*iv2, mu'Wmu, -2ln(a)]
//   featD[p] =       [ 1,   1,   1,  -2mu0,    -2mu1,    -2mu2,      |mu|^2,   0   ]
//   phs2[p]  = phases[p] / (2*pi)            (phase kept in "turns" for raw v_sin/cos)
// Point features (main kernel): g = [x0^2,x1^2,x2^2,x0,x1,x2,1,1]
//   exp2(featM.g) = a * exp(-0.5*maha)       (clamp dropped: maha >= 0 up to rounding)
//   featD.g = d2 (exact); sqrt(|d2|) guards against rounding residue for free
// ---------------------------------------------------------------------------
__global__ __launch_bounds__(256) void cg_prim_features(
    const float* __restrict__ pos, const float* __restrict__ scl,
    const float* __restrict__ amp, const float* __restrict__ phs,
    float* __restrict__ featM, float* __restrict__ featD,
    float* __restrict__ phs2)
{
    int p = blockIdx.x * 256 + threadIdx.x;
    if (p >= N_PRIMS) return;
    float mx = pos[3*p+0], my = pos[3*p+1], mz = pos[3*p+2];
    float sx = scl[3*p+0], sy = scl[3*p+1], sz = scl[3*p+2];
    float ivx = 1.0f/(sx*sx), ivy = 1.0f/(sy*sy), ivz = 1.0f/(sz*sz);
    float mu2w = mx*mx*ivx + my*my*ivy + mz*mz*ivz;
    float mu2  = mx*mx + my*my + mz*mz;
    float a    = amp[p];
    float fold = -2.0f * __logf(a);      // a==0 -> +inf -> lam*inf=-inf -> exp2=0

    const float lam = NEG_HALF_LOG2E;
    float* fm = featM + 8*p;
    fm[0] = lam*ivx;            fm[1] = lam*ivy;            fm[2] = lam*ivz;
    fm[3] = lam*-2.0f*mx*ivx;   fm[4] = lam*-2.0f*my*ivy;   fm[5] = lam*-2.0f*mz*ivz;
    fm[6] = lam*mu2w;           fm[7] = lam*fold;

    float* fd = featD + 8*p;
    fd[0] = 1.0f;     fd[1] = 1.0f;     fd[2] = 1.0f;
    fd[3] = -2.0f*mx; fd[4] = -2.0f*my; fd[5] = -2.0f*mz;
    fd[6] = mu2;      fd[7] = 0.0f;

    phs2[p] = phs[p] * 0.15915494309189535f;     // 1/(2*pi)
}

// ---------------------------------------------------------------------------
// Main: one wave32 per 16-point block. A = 16 primitives (M), B = 16 points (N),
// D(16x16) via chained V_WMMA_F32_16X16X4_F32 (K=8 as 2x K=4).
// A layout (32-bit 16x4): lanes 0-15 hold M=lane {K0,K1}; lanes 16-31 M=lane-16 {K2,K3}.
// B layout mirrored:      lanes 0-15 hold N=lane {K0,K1}; lanes 16-31 N=lane-16 {K2,K3}.
// D layout: lane holds column N=lane%16, rows M = i + 8*(lane>=16) in v8f slot i.
// ---------------------------------------------------------------------------
__global__ __launch_bounds__(256) void cg_render(
    const float* __restrict__ qp,
    const float* __restrict__ featM, const float* __restrict__ featD,
    const float* __restrict__ phs2,
    const unsigned* __restrict__ freq,
    float* __restrict__ out)
{
    const int lane = threadIdx.x & 31;
    const int wave = threadIdx.x >> 5;
    const int tile = blockIdx.x * 8 + wave;      // 16-point block, 0..2047
    const int hi   = lane >> 4;                  // half-wave select
    const int lm   = lane & 15;
    const int n    = tile * 16 + lm;             // this lane's point (column N)

    const float f   = (float)(*freq);
    const float kw2 = f / 299792458.0f;          // wavenumber in turns/metre: k/(2*pi)

    // Point features g = [x0^2,x1^2,x2^2,x0,x1,x2,1,1], split into K-chunks per layout.
    const float x0 = qp[3*n+0], x1 = qp[3*n+1], x2 = qp[3*n+2];
    v2f b0, b1;                                  // B chunk0 (K0..3), chunk1 (K4..7)
    b0.x = hi ? x2*x2 : x0*x0;                   // K2 : K0
    b0.y = hi ? x0    : x1*x1;                   // K3 : K1
    b1.x = hi ? 1.0f  : x1;                      // K6 : K4
    b1.y = hi ? 1.0f  : x2;                      // K7 : K5

    const int koff = hi * 2;                     // A: lo lanes K{0,1}/K{4,5}, hi lanes K{2,3}/K{6,7}
    float accRe = 0.0f, accIm = 0.0f;

    for (int t = 0; t < N_PRIMS / 16; ++t) {
        const int pbase = t * 16;

        // A operands: this lane supplies row M = lm (primitive pbase+lm).
        const float* fm = featM + 8 * (pbase + lm);
        const float* fd = featD + 8 * (pbase + lm);
        v2f aM0 = *(const v2f*)(fm + koff);
        v2f aM1 = *(const v2f*)(fm + 4 + koff);
        v2f aD0 = *(const v2f*)(fd + koff);
        v2f aD1 = *(const v2f*)(fd + 4 + koff);

        v8f lm8 = {};   // lam * (maha - 2 ln a): ready-to-use exp2 argument
        lm8 = __builtin_amdgcn_wmma_f32_16x16x4_f32(false, aM0, false, b0,
                                                    (short)0, lm8, false, false);
        lm8 = __builtin_amdgcn_wmma_f32_16x16x4_f32(false, aM1, false, b1,
                                                    (short)0, lm8, false, false);
        v8f d2 = {};
        d2 = __builtin_amdgcn_wmma_f32_16x16x4_f32(false, aD0, false, b0,
                                                   (short)0, d2, false, false);
        d2 = __builtin_amdgcn_wmma_f32_16x16x4_f32(false, aD1, false, b1,
                                                   (short)0, d2, false, false);

        // Element i corresponds to primitive pbase + i + 8*hi (contiguous 8).
        const float4 ph0 = *(const float4*)(phs2 + pbase + 8*hi);
        const float4 ph1 = *(const float4*)(phs2 + pbase + 8*hi + 4);
        float phv[8] = {ph0.x, ph0.y, ph0.z, ph0.w, ph1.x, ph1.y, ph1.z, ph1.w};

#pragma unroll
        for (int i = 0; i < 8; ++i) {
            float w   = __builtin_amdgcn_exp2f(lm8[i]);                   // v_exp_f32
            float r   = __builtin_amdgcn_sqrtf(__builtin_fabsf(d2[i]));   // v_sqrt_f32 |src|
            float phf = __builtin_fmaf(kw2, r, phv[i]);                   // phase in turns
            accRe = __builtin_fmaf(w, __builtin_amdgcn_cosf(phf), accRe); // raw v_cos_f32
            accIm = __builtin_fmaf(w, __builtin_amdgcn_sinf(phf), accIm); // raw v_sin_f32
        }
    }

    // Combine the two half-wave partial sums (same point column N=lm).
    accRe += __shfl_xor(accRe, 16, 32);
    accIm += __shfl_xor(accIm, 16, 32);
    if (hi == 0) {
        out[2*n + 0] = accRe;   // complex64: interleaved re/im
        out[2*n + 1] = accIm;
    }
}

// ---------------------------------------------------------------------------
extern "C" void kernel_launch(void* const* d_in, const int* in_sizes, int n_in,
                              void* d_out, int out_size, void* d_ws, size_t ws_size,
                              hipStream_t stream)
{
    const float*    qp   = (const float*)d_in[0];    // [32768,3]
    const float*    pos  = (const float*)d_in[1];    // [2048,3]
    const float*    scl  = (const float*)d_in[2];    // [2048,3]
    const float*    amp  = (const float*)d_in[3];    // [2048]
    const float*    phs  = (const float*)d_in[4];    // [2048]
    const unsigned* freq = (const unsigned*)d_in[5]; // 2400000000 fits u32 (low word if i64)
    float* out = (float*)d_out;

    float* featM = (float*)d_ws;                 // [2048][8]
    float* featD = featM + N_PRIMS * 8;          // [2048][8]
    float* phs2  = featD + N_PRIMS * 8;          // [2048]   (136 KB total in d_ws)

    cg_prim_features<<<N_PRIMS / 256, 256, 0, stream>>>(pos, scl, amp, phs,
                                                        featM, featD, phs2);

    // 32768/16 = 2048 waves; 8 waves (256 threads) per block -> 256 blocks.
    cg_render<<<N_POINTS / 16 / 8, 256, 0, stream>>>(qp, featM, featD, phs2, freq, out);
}